// AttEdgeAwareGCN_28312424415407
// MI455X (gfx1250) — compile-verified
//
#include <hip/hip_runtime.h>
#include <hip/hip_bf16.h>

// ---------------------------------------------------------------------------
// AttEdgeAwareGCN forward for MI455X (gfx1250), f32 end-to-end.
// All GEMMs via V_WMMA_F32_16X16X4_F32 (exact f32 matrix pipe).
// Each wave computes 5 M-tiles x 1 N-tile so the B (weight) fragment is
// loaded once per k-step and reused by 5 WMMAs (3125 and 50000 tiles are
// both divisible by 5 -> exact grids, no EXEC-divergent bounds checks).
// ---------------------------------------------------------------------------

#define N_NODES   50000
#define N_EDGES   800000
#define NODE_DIM  128
#define EDGE_DIM  64
#define HIDDEN    128
#define OUT_DIM   64
#define MT        5   // M-tiles per wave

typedef __attribute__((ext_vector_type(2))) float v2f;
typedef __attribute__((ext_vector_type(8))) float v8f;

__device__ __forceinline__ v8f wmma4(v2f a, v2f b, v8f c) {
  // D = A(16x4 f32) * B(4x16 f32) + C(16x16 f32)
  return __builtin_amdgcn_wmma_f32_16x16x4_f32(
      /*neg_a=*/false, a, /*neg_b=*/false, b,
      /*c_mod=*/(short)0, c, /*reuse_a=*/false, /*reuse_b=*/false);
}

// -------------------------------- utility ----------------------------------

__global__ void zero_kernel(float* __restrict__ p, int n) {
  for (int i = blockIdx.x * blockDim.x + threadIdx.x; i < n;
       i += gridDim.x * blockDim.x)
    p[i] = 0.0f;
}

// ---------------------- degree + sage neighbor sum -------------------------
// deg[c] += 1 per edge; s[col] += edge_features[row] (64 floats).
__global__ void deg_s_kernel(const int* __restrict__ row, const int* __restrict__ col,
                             const float* __restrict__ ef,
                             float* __restrict__ deg, float* __restrict__ s) {
  int g = blockIdx.x * blockDim.x + threadIdx.x;
  int j = g >> 4, q = g & 15;
  if (j >= N_EDGES) return;
  int r = row[j], c = col[j];
  float4 v = ((const float4*)ef)[r * 16 + q];
  float* sd = s + (size_t)c * 64 + q * 4;
  atomicAdd(sd + 0, v.x); atomicAdd(sd + 1, v.y);
  atomicAdd(sd + 2, v.z); atomicAdd(sd + 3, v.w);
  if (q == 0) atomicAdd(&deg[c], 1.0f);
}

__global__ void dinv_kernel(const float* __restrict__ deg, float* __restrict__ dinv) {
  int i = blockIdx.x * blockDim.x + threadIdx.x;
  if (i < N_NODES) dinv[i] = rsqrtf(deg[i] + 1.0f);
}

// ------------------------- generic f32 WMMA GEMM ---------------------------
// H[m0:m0+80, n0:n0+16] = X * W ; one wave, 5 M-tiles, B reused 5x.
__global__ void gemm_wmma_f32(const float* __restrict__ X, const float* __restrict__ W,
                              float* __restrict__ H, int K, int ldx, int ldw, int ldh) {
  int m0 = blockIdx.x * (16 * MT), n0 = blockIdx.y * 16;
  int lane = threadIdx.x, half = lane >> 4, mr = lane & 15;
  v8f c[MT] = {};
  for (int k = 0; k < K; k += 4) {
    int ka = k + half * 2;
    v2f b;
    b.x = W[(size_t)ka * ldw + n0 + mr];
    b.y = W[(size_t)(ka + 1) * ldw + n0 + mr];
#pragma unroll
    for (int t = 0; t < MT; ++t) {
      v2f a;
      const float* xr = X + (size_t)(m0 + t * 16 + mr) * ldx + ka;
      a.x = xr[0];
      a.y = xr[1];
      c[t] = wmma4(a, b, c[t]);
    }
  }
#pragma unroll
  for (int t = 0; t < MT; ++t)
#pragma unroll
    for (int v = 0; v < 8; ++v)
      H[(size_t)(m0 + t * 16 + v + half * 8) * ldh + n0 + mr] = c[t][v];
}

// ------------------------------ GCN agg ------------------------------------
// agg[col] += dinv[row]*dinv[col] * h[row]  (128 floats per edge)
__global__ void gcn_agg_kernel(const int* __restrict__ row, const int* __restrict__ col,
                               const float* __restrict__ dinv, const float* __restrict__ h,
                               float* __restrict__ agg) {
  int g = blockIdx.x * blockDim.x + threadIdx.x;
  int j = g >> 5, q = g & 31;
  if (j >= N_EDGES) return;
  int r = row[j], c = col[j];
  float norm = dinv[r] * dinv[c];
  float4 v = ((const float4*)h)[(size_t)r * 32 + q];
  float* a = agg + (size_t)c * 128 + q * 4;
  atomicAdd(a + 0, norm * v.x); atomicAdd(a + 1, norm * v.y);
  atomicAdd(a + 2, norm * v.z); atomicAdd(a + 3, norm * v.w);
}

// x = relu(agg + dinv^2 * h + b)
__global__ void gcn_fin_kernel(const float* __restrict__ h, const float* __restrict__ agg,
                               const float* __restrict__ dinv, const float* __restrict__ bias,
                               float* __restrict__ x) {
  int g = blockIdx.x * blockDim.x + threadIdx.x;
  if (g >= N_NODES * 128) return;
  int i = g >> 7, d = g & 127;
  float di = dinv[i];
  float v = agg[g] + di * di * h[g] + bias[d];
  x[g] = v > 0.0f ? v : 0.0f;
}

// mean = s / max(cnt,1), in place (cnt == raw degree)
__global__ void mean_kernel(float* __restrict__ s, const float* __restrict__ deg) {
  int g = blockIdx.x * blockDim.x + threadIdx.x;
  if (g >= N_NODES * 64) return;
  float c = deg[g >> 6];
  c = c < 1.0f ? 1.0f : c;
  s[g] = s[g] / c;
}

// --------------------- attention scores (fused GEMM) -----------------------
// scores[j] = leaky_relu( sum_n (ef[j]*W_att + b_att)_n^2 ); 80 edges per wave
__global__ void scores_kernel(const float* __restrict__ ef, const float* __restrict__ Watt,
                              const float* __restrict__ batt, float* __restrict__ scores) {
  int e0 = blockIdx.x * (16 * MT);
  int lane = threadIdx.x, half = lane >> 4, mr = lane & 15;
  float sq[MT][8] = {};
  for (int ch = 0; ch < 8; ++ch) {
    int n0 = ch * 16;
    v8f c[MT] = {};
    for (int k = 0; k < 64; k += 4) {
      int ka = k + half * 2;
      v2f b;
      b.x = Watt[(size_t)ka * 128 + n0 + mr];
      b.y = Watt[(size_t)(ka + 1) * 128 + n0 + mr];
#pragma unroll
      for (int t = 0; t < MT; ++t) {
        v2f a;
        const float* xr = ef + (size_t)(e0 + t * 16 + mr) * 64 + ka;
        a.x = xr[0];
        a.y = xr[1];
        c[t] = wmma4(a, b, c[t]);
      }
    }
    float bb = batt[n0 + mr];
#pragma unroll
    for (int t = 0; t < MT; ++t)
#pragma unroll
      for (int v = 0; v < 8; ++v) {
        float e = c[t][v] + bb;
        sq[t][v] += e * e;
      }
  }
  // reduce across the 16-lane half (cols); rows stay per (reg, half)
#pragma unroll
  for (int t = 0; t < MT; ++t)
#pragma unroll
    for (int v = 0; v < 8; ++v) {
      float s = sq[t][v];
      s += __shfl_xor(s, 1);
      s += __shfl_xor(s, 2);
      s += __shfl_xor(s, 4);
      s += __shfl_xor(s, 8);
      sq[t][v] = s;
    }
  if (mr == 0) {
#pragma unroll
    for (int t = 0; t < MT; ++t)
#pragma unroll
      for (int v = 0; v < 8; ++v) {
        float s = sq[t][v];
        s = s < 0.0f ? 0.01f * s : s;  // leaky_relu (scores >= 0 anyway)
        scores[e0 + t * 16 + v + half * 8] = s;
      }
  }
}

// ------------------------- softmax reductions ------------------------------
__global__ void reduce_max_partial(const float* __restrict__ s, int n, float* __restrict__ part) {
  __shared__ float sm[256];
  float m = -3.4e38f;
  for (int i = blockIdx.x * blockDim.x + threadIdx.x; i < n; i += gridDim.x * blockDim.x)
    m = fmaxf(m, s[i]);
  sm[threadIdx.x] = m; __syncthreads();
  for (int o = 128; o > 0; o >>= 1) {
    if (threadIdx.x < o) sm[threadIdx.x] = fmaxf(sm[threadIdx.x], sm[threadIdx.x + o]);
    __syncthreads();
  }
  if (threadIdx.x == 0) part[blockIdx.x] = sm[0];
}

__global__ void reduce_max_final(const float* __restrict__ part, int n, float* __restrict__ red) {
  __shared__ float sm[256];
  float m = -3.4e38f;
  for (int i = threadIdx.x; i < n; i += 256) m = fmaxf(m, part[i]);
  sm[threadIdx.x] = m; __syncthreads();
  for (int o = 128; o > 0; o >>= 1) {
    if (threadIdx.x < o) sm[threadIdx.x] = fmaxf(sm[threadIdx.x], sm[threadIdx.x + o]);
    __syncthreads();
  }
  if (threadIdx.x == 0) red[0] = sm[0];
}

__global__ void reduce_sum_partial(const float* __restrict__ s, int n,
                                   const float* __restrict__ red, float* __restrict__ part) {
  __shared__ float sm[256];
  float gmax = red[0];
  float acc = 0.0f;
  for (int i = blockIdx.x * blockDim.x + threadIdx.x; i < n; i += gridDim.x * blockDim.x)
    acc += __expf(s[i] - gmax);
  sm[threadIdx.x] = acc; __syncthreads();
  for (int o = 128; o > 0; o >>= 1) {
    if (threadIdx.x < o) sm[threadIdx.x] += sm[threadIdx.x + o];
    __syncthreads();
  }
  if (threadIdx.x == 0) part[blockIdx.x] = sm[0];
}

__global__ void reduce_sum_final(const float* __restrict__ part, int n, float* __restrict__ red) {
  __shared__ float sm[256];
  float acc = 0.0f;
  for (int i = threadIdx.x; i < n; i += 256) acc += part[i];
  sm[threadIdx.x] = acc; __syncthreads();
  for (int o = 128; o > 0; o >>= 1) {
    if (threadIdx.x < o) sm[threadIdx.x] += sm[threadIdx.x + o];
    __syncthreads();
  }
  if (threadIdx.x == 0) red[1] = sm[0];
}

// ----------------------- agg_nbr scatter (edges) ---------------------------
// agg_nbr[col] += coeff[j] * x2[row[j]]
__global__ void agg_nbr_kernel(const int* __restrict__ row, const int* __restrict__ col,
                               const float* __restrict__ x2, const float* __restrict__ scores,
                               const float* __restrict__ red, float* __restrict__ agg) {
  int g = blockIdx.x * blockDim.x + threadIdx.x;
  int j = g >> 5, q = g & 31;
  if (j >= N_EDGES) return;
  int r = row[j], c = col[j];
  float coeff = __expf(scores[j] - red[0]) * (1.0f / red[1]);
  float4 v = ((const float4*)x2)[(size_t)r * 32 + q];
  float* a = agg + (size_t)c * 128 + q * 4;
  atomicAdd(a + 0, coeff * v.x); atomicAdd(a + 1, coeff * v.y);
  atomicAdd(a + 2, coeff * v.z); atomicAdd(a + 3, coeff * v.w);
}

// --------- sage edge feature + attention scatter (fused WMMA) --------------
// e[j] = relu(ef[j]*W_root + (j<N ? meanW[j] : 0) + b_sage)
// agg_edge[col[j]] += coeff[j] * e[j]; scattered straight from WMMA regs.
__global__ void edge_e_kernel(const float* __restrict__ ef, const float* __restrict__ Wroot,
                              const float* __restrict__ bs, const float* __restrict__ meanW,
                              const float* __restrict__ scores, const float* __restrict__ red,
                              const int* __restrict__ col, float* __restrict__ agg_edge) {
  int e0 = blockIdx.x * (16 * MT);
  int lane = threadIdx.x, half = lane >> 4, mr = lane & 15;
  float gmax = red[0];
  float rinv = 1.0f / red[1];
  int ecol[MT][8];
  float ecoef[MT][8];
#pragma unroll
  for (int t = 0; t < MT; ++t)
#pragma unroll
    for (int v = 0; v < 8; ++v) {
      int e = e0 + t * 16 + v + half * 8;
      ecol[t][v] = col[e];
      ecoef[t][v] = __expf(scores[e] - gmax) * rinv;
    }
  for (int ch = 0; ch < 8; ++ch) {
    int n0 = ch * 16;
    v8f c[MT] = {};
    for (int k = 0; k < 64; k += 4) {
      int ka = k + half * 2;
      v2f b;
      b.x = Wroot[(size_t)ka * 128 + n0 + mr];
      b.y = Wroot[(size_t)(ka + 1) * 128 + n0 + mr];
#pragma unroll
      for (int t = 0; t < MT; ++t) {
        v2f a;
        const float* xr = ef + (size_t)(e0 + t * 16 + mr) * 64 + ka;
        a.x = xr[0];
        a.y = xr[1];
        c[t] = wmma4(a, b, c[t]);
      }
    }
    int coln = n0 + mr;
    float bb = bs[coln];
#pragma unroll
    for (int t = 0; t < MT; ++t)
#pragma unroll
      for (int v = 0; v < 8; ++v) {
        int e = e0 + t * 16 + v + half * 8;
        float val = c[t][v] + bb;
        if (e < N_NODES) val += meanW[(size_t)e * 128 + coln];
        val = val > 0.0f ? val : 0.0f;
        atomicAdd(&agg_edge[(size_t)ecol[t][v] * 128 + coln], ecoef[t][v] * val);
      }
  }
}

// ------------------------- final output GEMM -------------------------------
// out = concat(x2 + agg_nbr, agg_edge) @ W_fc + b_fc   ([N,256]x[256,64])
__global__ void out_gemm_kernel(const float* __restrict__ x2, const float* __restrict__ agg_nbr,
                                const float* __restrict__ agg_edge, const float* __restrict__ Wfc,
                                const float* __restrict__ bfc, float* __restrict__ out) {
  int m0 = blockIdx.x * (16 * MT), n0 = blockIdx.y * 16;
  int lane = threadIdx.x, half = lane >> 4, mr = lane & 15;
  v8f c[MT] = {};
  for (int k = 0; k < 256; k += 4) {
    int ka = k + half * 2;
    v2f b;
    b.x = Wfc[(size_t)ka * 64 + n0 + mr];
    b.y = Wfc[(size_t)(ka + 1) * 64 + n0 + mr];
#pragma unroll
    for (int t = 0; t < MT; ++t) {
      int r = m0 + t * 16 + mr;
      v2f a;
      if (ka < 128) {  // uniform per k-step (k multiple of 4)
        a.x = x2[(size_t)r * 128 + ka] + agg_nbr[(size_t)r * 128 + ka];
        a.y = x2[(size_t)r * 128 + ka + 1] + agg_nbr[(size_t)r * 128 + ka + 1];
      } else {
        a.x = agg_edge[(size_t)r * 128 + ka - 128];
        a.y = agg_edge[(size_t)r * 128 + ka - 127];
      }
      c[t] = wmma4(a, b, c[t]);
    }
  }
#pragma unroll
  for (int t = 0; t < MT; ++t)
#pragma unroll
    for (int v = 0; v < 8; ++v)
      out[(size_t)(m0 + t * 16 + v + half * 8) * 64 + n0 + mr] = c[t][v] + bfc[n0 + mr];
}

// ---------------------------------------------------------------------------

extern "C" void kernel_launch(void* const* d_in, const int* in_sizes, int n_in,
                              void* d_out, int out_size, void* d_ws, size_t ws_size,
                              hipStream_t stream) {
  const float* node_feat = (const float*)d_in[0];
  const float* edge_feat = (const float*)d_in[1];
  const float* W_gc1     = (const float*)d_in[2];
  const float* b_gc1     = (const float*)d_in[3];
  const float* W_gc2     = (const float*)d_in[4];
  const float* b_gc2     = (const float*)d_in[5];
  const float* W_nbr     = (const float*)d_in[6];
  const float* W_root    = (const float*)d_in[7];
  const float* b_sage    = (const float*)d_in[8];
  const float* W_att     = (const float*)d_in[9];
  const float* b_att     = (const float*)d_in[10];
  const float* W_fc      = (const float*)d_in[11];
  const float* b_fc      = (const float*)d_in[12];
  const int*   row       = (const int*)d_in[13];
  const int*   col       = row + N_EDGES;
  float*       out       = (float*)d_out;

  // workspace carve-up (floats)
  float* ws      = (float*)d_ws;
  float* deg     = ws;                              // N
  float* dinv    = deg + N_NODES;                   // N
  float* bufA    = dinv + N_NODES;                  // N*128 (h1/h2, then meanW)
  float* bufB    = bufA + (size_t)N_NODES * 128;    // N*128 (x1/x2)
  float* bufC    = bufB + (size_t)N_NODES * 128;    // N*128 (agg, then agg_nbr)
  float* bufD    = bufC + (size_t)N_NODES * 128;    // N*128 (agg_edge)
  float* sbuf    = bufD + (size_t)N_NODES * 128;    // N*64  (sage sum -> mean)
  float* scores  = sbuf + (size_t)N_NODES * 64;     // E
  float* part    = scores + N_EDGES;                // 512
  float* red     = part + 512;                      // 2

  const dim3 b256(256), b32(32);
  const int MTILES = N_NODES / (16 * MT);  // 625 (exact)
  const int ETILES = N_EDGES / (16 * MT);  // 10000 (exact)

  // 1) degree + sage neighbor sums
  zero_kernel<<<2048, b256, 0, stream>>>(deg, N_NODES);
  zero_kernel<<<2048, b256, 0, stream>>>(sbuf, N_NODES * 64);
  deg_s_kernel<<<(N_EDGES * 16) / 256, b256, 0, stream>>>(row, col, edge_feat, deg, sbuf);
  dinv_kernel<<<(N_NODES + 255) / 256, b256, 0, stream>>>(deg, dinv);

  // 2) GCN layer 1
  gemm_wmma_f32<<<dim3(MTILES, 8), b32, 0, stream>>>(node_feat, W_gc1, bufA, 128, 128, 128, 128);
  zero_kernel<<<2048, b256, 0, stream>>>(bufC, N_NODES * 128);
  gcn_agg_kernel<<<(N_EDGES * 32) / 256, b256, 0, stream>>>(row, col, dinv, bufA, bufC);
  gcn_fin_kernel<<<(N_NODES * 128) / 256, b256, 0, stream>>>(bufA, bufC, dinv, b_gc1, bufB);

  // 3) GCN layer 2
  gemm_wmma_f32<<<dim3(MTILES, 8), b32, 0, stream>>>(bufB, W_gc2, bufA, 128, 128, 128, 128);
  zero_kernel<<<2048, b256, 0, stream>>>(bufC, N_NODES * 128);
  gcn_agg_kernel<<<(N_EDGES * 32) / 256, b256, 0, stream>>>(row, col, dinv, bufA, bufC);
  gcn_fin_kernel<<<(N_NODES * 128) / 256, b256, 0, stream>>>(bufA, bufC, dinv, b_gc2, bufB);  // x2

  // 4) sage mean + meanW = mean @ W_sage_nbr   (bufA reused as meanW)
  mean_kernel<<<(N_NODES * 64) / 256, b256, 0, stream>>>(sbuf, deg);
  gemm_wmma_f32<<<dim3(MTILES, 8), b32, 0, stream>>>(sbuf, W_nbr, bufA, 64, 64, 128, 128);

  // 5) attention scores + global softmax stats
  scores_kernel<<<ETILES, b32, 0, stream>>>(edge_feat, W_att, b_att, scores);
  reduce_max_partial<<<512, b256, 0, stream>>>(scores, N_EDGES, part);
  reduce_max_final<<<1, b256, 0, stream>>>(part, 512, red);
  reduce_sum_partial<<<512, b256, 0, stream>>>(scores, N_EDGES, red, part);
  reduce_sum_final<<<1, b256, 0, stream>>>(part, 512, red);

  // 6) attention-weighted aggregations
  zero_kernel<<<2048, b256, 0, stream>>>(bufC, N_NODES * 128);  // agg_nbr
  zero_kernel<<<2048, b256, 0, stream>>>(bufD, N_NODES * 128);  // agg_edge
  agg_nbr_kernel<<<(N_EDGES * 32) / 256, b256, 0, stream>>>(row, col, bufB, scores, red, bufC);
  edge_e_kernel<<<ETILES, b32, 0, stream>>>(edge_feat, W_root, b_sage, bufA, scores, red, col, bufD);

  // 7) output projection
  out_gemm_kernel<<<dim3(MTILES, OUT_DIM / 16), b32, 0, stream>>>(bufB, bufC, bufD, W_fc, b_fc, out);
}